// CTCLayer_12103217840723
// MI455X (gfx1250) — compile-verified
//
#include <hip/hip_runtime.h>
#include <stdint.h>

// ---------------------------------------------------------------------------
// CTC loss, fused single-pass kernel for MI455X (gfx1250).
//   B=256, T=1024, C=256, L=128, S=2L+1=257, blank = C-1.
// One workgroup per batch row: 288 threads (9 wave32), lane s<257 owns state s.
// y_pred[b] is streamed HBM->LDS with the Tensor Data Mover in 16-timestep
// chunks (double buffered); per-chunk softmax denominators are computed on the
// matrix pipe with V_WMMA_F32_16X16X4_F32 (ones-matmul row sums, exact f32).
// All log-prob math is done in log2 domain (v_log_f32 / v_exp_f32 are single
// transcendental instructions); the recursion is scale invariant, final *ln2.
// ---------------------------------------------------------------------------

typedef __attribute__((ext_vector_type(2))) float        v2f;
typedef __attribute__((ext_vector_type(8))) float        v8f;
typedef __attribute__((ext_vector_type(4))) unsigned int v4u;
typedef __attribute__((ext_vector_type(4))) int          v4i;
typedef __attribute__((ext_vector_type(8))) int          v8i;

#define CTC_B   256
#define CTC_T   1024
#define CTC_C   256
#define CTC_L   128
#define CTC_S   257            // 2L+1
#define CHUNK   16             // timesteps staged per TDM tile
#define NCHUNK  (CTC_T / CHUNK)
#define NEGF    (-1.0e30f)
#define EPSF    (1.0e-7f)
#define LN2F    (0.69314718055994530942f)

// 32-bit LDS offset of a __shared__ object (flat LDS aperture keeps the
// wave-relative byte offset in addr[31:0], per CDNA5 ISA 10.2).
__device__ __forceinline__ unsigned ldsOff(const void* p) {
  return (unsigned)(uintptr_t)p;
}

// Issue a TDM load of a [CHUNK x CTC_C] fp32 tile (row-major, contiguous rows)
// from global `gaddr` into LDS byte offset `lds`. D# packing per ISA ch.8.
// This toolchain's builtin takes 6 args (4 descriptor groups, one extra
// 8-dword group, cache policy).
__device__ __forceinline__ void tdm_load_chunk(uint64_t gaddr, unsigned lds) {
  v4u g0;
  g0.x = 1u;                                      // count=1, user descriptor
  g0.y = lds;                                     // lds_addr (bytes)
  g0.z = (unsigned)(gaddr & 0xffffffffu);         // global_addr[31:0]
  g0.w = (unsigned)(gaddr >> 32) | (2u << 30);    // global_addr[56:32] | type=2
  v8i g1;
  g1[0] = 0x00020000;                   // data_size=2 (4B); no multicast/pad
  g1[1] = (CTC_C & 0xffff) << 16;       // tensor_dim0 = 256   (bits 63:48)
  g1[2] = (CHUNK) << 16;                // tensor_dim1 = CHUNK (bits 95:80)
  g1[3] = (CTC_C & 0xffff) << 16;       // tile_dim0   = 256   (bits 127:112)
  g1[4] = CHUNK;                        // tile_dim1   = CHUNK (bits 143:128)
  g1[5] = CTC_C;                        // tensor_dim0_stride  = 256
  g1[6] = 0;
  g1[7] = 0;
  v4i g2 = {0, 0, 0, 0};
  v4i g3 = {0, 0, 0, 0};
  v8i g4 = {0, 0, 0, 0, 0, 0, 0, 0};    // unused extra group (<=2D tensor)
  __builtin_amdgcn_tensor_load_to_lds(g0, g1, g2, g3, g4, 0);
}

// 3-way log-sum-exp in log2 domain: m + log2(2^(a1-m)+2^(a2-m)+2^(a3-m)).
__device__ __forceinline__ float lse3_log2(float a1, float a2, float a3) {
  float m = fmaxf(fmaxf(a1, a2), a3);
  float s = __builtin_amdgcn_exp2f(a1 - m) +
            __builtin_amdgcn_exp2f(a2 - m) +
            __builtin_amdgcn_exp2f(a3 - m);
  return m + __builtin_amdgcn_logf(s);            // v_log_f32 == log2
}

__global__ __launch_bounds__(288, 2)
void ctc_fused_kernel(const float* __restrict__ y_pred,
                      const int*   __restrict__ y_true,
                      float*       __restrict__ out)
{
  __shared__ float pbuf[2][CHUNK][CTC_C];   // 32 KB double-buffered prob tiles
  __shared__ float Abuf[2][CTC_S + 2];      // alpha ping-pong, 2 NEG pad slots
  __shared__ float dlog[CHUNK];             // log2 of per-timestep denominators

  const int  b       = blockIdx.x;
  const int  tid     = threadIdx.x;
  const bool isState = (tid < CTC_S);

  // Per-state constants: class index and skip-transition permission.
  int  cls  = CTC_C - 1;                    // blank
  bool skip = false;
  if (isState && (tid & 1)) {               // odd state -> label (tid-1)/2
    const int li = tid >> 1;
    cls  = y_true[b * CTC_L + li];
    skip = (tid < 3) || (cls != y_true[b * CTC_L + li - 1]);
  }

  // Init alpha buffers (state slots and the two -inf pads).
  if (isState) { Abuf[0][2 + tid] = NEGF; Abuf[1][2 + tid] = NEGF; }
  if (tid < 2) { Abuf[0][tid] = NEGF;     Abuf[1][tid] = NEGF; }

  // Wave-scalar "am I wave 0" (forces an s_cbranch so WMMA runs with EXEC=~0
  // and the TDM op is issued by exactly one wave).
  const bool isw0 = (__builtin_amdgcn_readfirstlane((int)(tid >> 5)) == 0);

  const uint64_t gbase = (uint64_t)(uintptr_t)y_pred +
                         (uint64_t)b * CTC_T * CTC_C * sizeof(float);

  if (isw0) tdm_load_chunk(gbase, ldsOff(&pbuf[0][0][0]));   // prefetch chunk 0

  int pb = 0;                                                // alpha read buffer
  for (int c = 0; c < NCHUNK; ++c) {
    const int cur = c & 1;
    const int nxt = cur ^ 1;

    if (isw0) __builtin_amdgcn_s_wait_tensorcnt(0);          // tile c arrived
    __syncthreads();                                         // visible to all

    if (isw0) {
      // Prefetch next tile while this chunk is consumed.
      if (c + 1 < NCHUNK)
        tdm_load_chunk(gbase + (uint64_t)(c + 1) * CHUNK * CTC_C * sizeof(float),
                       ldsOff(&pbuf[nxt][0][0]));

      // Row sums of the 16x256 tile on the matrix pipe:
      //   D = P(16x256) x ones(256x16), chained V_WMMA_F32_16X16X4_F32.
      // A layout (32b A 16x4): lanes 0-15 -> K=0,1 ; lanes 16-31 -> K=2,3.
      const float* pc   = &pbuf[cur][0][0];
      const int    lane = tid;                 // 0..31 inside wave 0
      const int    row  = lane & 15;           // M (timestep within chunk)
      const int    koff = (lane >> 4) << 1;    // 0 or 2
      const float* ap   = pc + (row << 8) + koff;
      v8f acc   = {0.f, 0.f, 0.f, 0.f, 0.f, 0.f, 0.f, 0.f};
      v2f bones = {1.f, 1.f};
      #pragma unroll 8
      for (int k = 0; k < CTC_C; k += 4) {
        v2f a;
        a.x = ap[k];
        a.y = ap[k + 1];
        acc = __builtin_amdgcn_wmma_f32_16x16x4_f32(
            false, a, false, bones, (short)0, acc, false, false);
      }
      // D 16x16 f32 layout: lanes 0-15 vgpr r -> M=r, lanes 16-31 -> M=8+r.
      if (row == 0) {
        const float* af    = (const float*)&acc;
        const int    mbase = (lane >> 4) << 3;   // 0 or 8
        #pragma unroll
        for (int r = 0; r < 8; ++r)
          dlog[mbase + r] = __builtin_amdgcn_logf(af[r] + CTC_C * EPSF);
      }
    }
    __syncthreads();                                         // dlog ready

    // 16 recursion steps on this chunk (1 barrier per step).
    for (int tt = 0; tt < CHUNK; ++tt) {
      const int t  = (c << 4) + tt;
      float     av = NEGF;
      if (isState) {
        const float lp =
            __builtin_amdgcn_logf(pbuf[cur][tt][cls] + EPSF) - dlog[tt];
        if (t == 0) {
          av = (tid < 2) ? lp : NEGF;
        } else {
          const float a1 = Abuf[pb][2 + tid];
          const float a2 = Abuf[pb][1 + tid];
          const float a3 = skip ? Abuf[pb][tid] : NEGF;
          av = lse3_log2(a1, a2, a3) + lp;
        }
      }
      const int nb = pb ^ 1;
      if (isState) Abuf[nb][2 + tid] = av;
      __syncthreads();
      pb = nb;
    }
  }

  // loss[b] = -ln2 * log2(2^alpha[S-1] + 2^alpha[S-2])
  if (tid == 0) {
    const float aN  = Abuf[pb][2 + CTC_S - 1];
    const float aN1 = Abuf[pb][2 + CTC_S - 2];
    const float m   = fmaxf(aN, aN1);
    const float l2  = m + __builtin_amdgcn_logf(
                              __builtin_amdgcn_exp2f(aN - m) +
                              __builtin_amdgcn_exp2f(aN1 - m));
    out[b] = -LN2F * l2;
  }
}

extern "C" void kernel_launch(void* const* d_in, const int* in_sizes, int n_in,
                              void* d_out, int out_size, void* d_ws,
                              size_t ws_size, hipStream_t stream) {
  (void)in_sizes; (void)n_in; (void)out_size; (void)d_ws; (void)ws_size;
  const int*   y_true = (const int*)d_in[0];    // [B, L] int labels
  const float* y_pred = (const float*)d_in[1];  // [B, T, C] fp32 probs
  float*       out    = (float*)d_out;          // [B, 1] fp32 loss
  hipLaunchKernelGGL(ctc_fused_kernel, dim3(CTC_B), dim3(288), 0, stream,
                     y_pred, y_true, out);
}